// VINeuralODE_57647051047068
// MI455X (gfx1250) — compile-verified
//
#include <hip/hip_runtime.h>
#include <hip/hip_bf16.h>

#define D 32
#define H 128
#define BATCH 128
#define NSTEPS 8
#define MT 16          // samples per block = WMMA M
#define LOG2PI 1.8378770664093453f

typedef float v2f __attribute__((ext_vector_type(2)));
typedef float v8f __attribute__((ext_vector_type(8)));

// DOPRI5 Butcher tableau (matches reference)
__constant__ float cA[6][5] = {
  {0.f, 0.f, 0.f, 0.f, 0.f},
  {0.2f, 0.f, 0.f, 0.f, 0.f},
  {3.f/40.f, 9.f/40.f, 0.f, 0.f, 0.f},
  {44.f/45.f, -56.f/15.f, 32.f/9.f, 0.f, 0.f},
  {19372.f/6561.f, -25360.f/2187.f, 64448.f/6561.f, -212.f/729.f, 0.f},
  {9017.f/3168.f, -355.f/33.f, 46732.f/5247.f, 49.f/176.f, -5103.f/18656.f}
};
__constant__ float cB[6] = {35.f/384.f, 0.f, 500.f/1113.f, 125.f/192.f, -2187.f/6784.f, 11.f/84.f};
__constant__ float cC[6] = {0.f, 0.2f, 0.3f, 0.8f, 8.f/9.f, 1.f};

__device__ __forceinline__ float fast_tanh(float x) {
  // tanh(x) = 1 - 2/(exp(2x)+1); v_exp_f32 + v_rcp_f32 (~1 ulp), saturates at +/-1
  float e = __builtin_amdgcn_exp2f(x * 2.8853900817779268f); // 2*log2(e)
  return 1.0f - 2.0f * __builtin_amdgcn_rcpf(e + 1.0f);
}

// P[k][l] = W2[k][l] * C[l][k],  C = W1 @ W3  (sample/t independent -> hoisted)
__global__ void vode_prep(const float* __restrict__ W1, const float* __restrict__ W2,
                          const float* __restrict__ W3, float* __restrict__ P) {
  int idx = blockIdx.x * blockDim.x + threadIdx.x;   // 0 .. H*H-1
  int k = idx >> 7;
  int l = idx & (H - 1);
  float c = 0.f;
  #pragma unroll
  for (int i = 0; i < D; ++i) c += W1[l * D + i] * W3[i * H + k];
  P[k * H + l] = W2[k * H + l] * c;
}

__global__ __launch_bounds__(256)
void vode_main(const float* __restrict__ x0,
               const float* __restrict__ W1, const float* __restrict__ u1,
               const float* __restrict__ b1, const float* __restrict__ W2,
               const float* __restrict__ b2, const float* __restrict__ W3,
               const float* __restrict__ b3, const float* __restrict__ prec,
               const float* __restrict__ P, float* __restrict__ out) {
  // LDS (padded strides chosen for conflict-free ds_load_b64 A-fragments)
  __shared__ float ycur[MT][36];
  __shared__ float ystage[MT][36];
  __shared__ float kbuf[6][MT][36];   // k-stage outputs: x[0..31], dlogp[32], dkl[33]
  __shared__ float h1v[MT][132];      // h1; reused as trace-product buffer
  __shared__ float d1v[MT][132];      // 1 - h1^2
  __shared__ float h2v[MT][132];      // h2
  __shared__ float d2v[MT][132];      // 1 - h2^2
  __shared__ float red[MT][17];

  const int tid  = threadIdx.x;
  const int wave = tid >> 5;
  const int lane = tid & 31;
  const int ln   = lane & 15;        // A-row m / B-col n
  const int hi   = lane >> 4;        // K-pair select
  const int kb   = hi * 2;
  const int col  = wave * 16 + ln;   // this wave's output column
  const int sbase = blockIdx.x * MT;

  // ---- B fragments held in registers for all 48 field evals ----
  v2f b1f[8], b2f[32], bpf[32];
  {
    const float* base = W1 + col * D + kb;        // B[k=d][n=h] = W1[h][d]
    #pragma unroll
    for (int s = 0; s < 8; ++s) b1f[s] = *(const v2f*)(base + 4 * s);
  }
  {
    const float* base = W2 + col * H + kb;        // B[k=l][n=k2] = W2[k2][l]
    #pragma unroll
    for (int s = 0; s < 32; ++s) b2f[s] = *(const v2f*)(base + 4 * s);
  }
  {
    const float* base = P + col * H + kb;         // B[k=l][n=k2] = P[k2][l]
    #pragma unroll
    for (int s = 0; s < 32; ++s) bpf[s] = *(const v2f*)(base + 4 * s);
  }
  const float u1c = u1[col];
  const float b1c = b1[col];
  const float b2c = b2[col];

  // init state: x from x0, logdet = kl = 0
  for (int idx = tid; idx < MT * 36; idx += 256) {
    int s = idx / 36, c = idx - s * 36;
    ycur[s][c] = (c < D) ? x0[(sbase + s) * D + c] : 0.f;
  }
  __syncthreads();

  const float dtv = 1.0f / NSTEPS;

  for (int step = 0; step < NSTEPS; ++step) {
    const float t0 = step * dtv;
    for (int st = 0; st < 6; ++st) {
      // ---- build stage state: ystage = y + dt * sum_j a[st][j] k_j ----
      for (int idx = tid; idx < MT * 34; idx += 256) {
        int s = idx / 34, c = idx - s * 34;
        float v = ycur[s][c];
        for (int j = 0; j < st; ++j) v += dtv * cA[st][j] * kbuf[j][s][c];
        ystage[s][c] = v;
      }
      __syncthreads();
      const float t = t0 + cC[st] * dtv;

      // ---- layer 1: (16x32) @ W1^T -> 16x128, tanh ----
      v8f acc = {0.f,0.f,0.f,0.f,0.f,0.f,0.f,0.f};
      #pragma unroll
      for (int s = 0; s < 8; ++s) {
        v2f a = *(const v2f*)&ystage[ln][4 * s + kb];
        acc = __builtin_amdgcn_wmma_f32_16x16x4_f32(false, a, false, b1f[s],
                                                    (short)0, acc, false, false);
      }
      {
        float ub = t * u1c + b1c;
        #pragma unroll
        for (int r = 0; r < 8; ++r) {
          int mm = r + 8 * hi;
          float h = fast_tanh(acc[r] + ub);
          h1v[mm][col] = h;
          d1v[mm][col] = 1.f - h * h;
        }
      }
      __syncthreads();

      // ---- layer 2: (16x128) @ W2^T -> 16x128, tanh ----
      acc = (v8f){0.f,0.f,0.f,0.f,0.f,0.f,0.f,0.f};
      #pragma unroll
      for (int s = 0; s < 32; ++s) {
        v2f a = *(const v2f*)&h1v[ln][4 * s + kb];
        acc = __builtin_amdgcn_wmma_f32_16x16x4_f32(false, a, false, b2f[s],
                                                    (short)0, acc, false, false);
      }
      #pragma unroll
      for (int r = 0; r < 8; ++r) {
        int mm = r + 8 * hi;
        float h = fast_tanh(acc[r] + b2c);
        h2v[mm][col] = h;
        d2v[mm][col] = 1.f - h * h;
      }
      __syncthreads();

      // ---- layer 3 (waves 0-1): dxdt = (16x128) @ W3^T + b3 -> kbuf ----
      if (wave < 2) {
        v8f c3 = {0.f,0.f,0.f,0.f,0.f,0.f,0.f,0.f};
        const float* base3 = W3 + col * H + kb;   // B[k=h][n=d] = W3[d][h]
        #pragma unroll
        for (int s = 0; s < 32; ++s) {
          v2f a = *(const v2f*)&h2v[ln][4 * s + kb];
          v2f b = *(const v2f*)(base3 + 4 * s);
          c3 = __builtin_amdgcn_wmma_f32_16x16x4_f32(false, a, false, b,
                                                     (short)0, c3, false, false);
        }
        float b3c = b3[col];
        #pragma unroll
        for (int r = 0; r < 8; ++r) kbuf[st][r + 8 * hi][col] = c3[r] + b3c;
      }
      // ---- trace: G = D1 @ P^T (16x128); prod = d2 .* G into h1v ----
      v8f ct = {0.f,0.f,0.f,0.f,0.f,0.f,0.f,0.f};
      #pragma unroll
      for (int s = 0; s < 32; ++s) {
        v2f a = *(const v2f*)&d1v[ln][4 * s + kb];
        ct = __builtin_amdgcn_wmma_f32_16x16x4_f32(false, a, false, bpf[s],
                                                   (short)0, ct, false, false);
      }
      #pragma unroll
      for (int r = 0; r < 8; ++r) {
        int mm = r + 8 * hi;
        h1v[mm][col] = ct[r] * d2v[mm][col];
      }
      __syncthreads();

      // ---- reduce trace + per-sample scalar terms ----
      {
        int s = tid >> 4, j = tid & 15;
        float p = 0.f;
        #pragma unroll
        for (int i = 0; i < 8; ++i) p += h1v[s][8 * j + i];
        red[s][j] = p;
      }
      __syncthreads();
      if (tid < MT) {
        float tr = 0.f;
        #pragma unroll
        for (int j = 0; j < 16; ++j) tr += red[tid][j];
        float dlogp = -tr;
        float a1 = 0.f, a2 = 0.f;
        for (int dd = 0; dd < D; ++dd) {
          float x  = ystage[tid][dd];
          float xd = kbuf[st][tid][dd];
          a1 += (-0.5f * x * x - 0.5f * LOG2PI) * xd;
          a2 += (-prec[dd] * x) * xd;
        }
        float omt = 1.f - t;
        kbuf[st][tid][32] = dlogp;
        kbuf[st][tid][33] = -0.5f * omt * omt * a1
                            - 0.5f * omt * (1.f + t) * a2 + omt * dlogp;
      }
      __syncthreads();
    } // stages

    // ---- y += dt * sum_j b[j] k_j ----
    for (int idx = tid; idx < MT * 34; idx += 256) {
      int s = idx / 34, c = idx - s * 34;
      float v = ycur[s][c];
      #pragma unroll
      for (int j = 0; j < 6; ++j) v += dtv * cB[j] * kbuf[j][s][c];
      ycur[s][c] = v;
    }
    __syncthreads();
  } // steps

  // ---- outputs: z (B,D) | log_px0+logdet (B) | kl (B) ----
  for (int idx = tid; idx < MT * D; idx += 256) {
    int s = idx / D, dd = idx - s * D;
    out[(sbase + s) * D + dd] = ycur[s][dd];
  }
  if (tid < MT) {
    float lp0 = 0.f;
    for (int dd = 0; dd < D; ++dd) {
      float x = x0[(sbase + tid) * D + dd];
      lp0 += -0.5f * x * x - 0.5f * LOG2PI;
    }
    out[BATCH * D + sbase + tid]         = lp0 + ycur[tid][32];
    out[BATCH * D + BATCH + sbase + tid] = ycur[tid][33];
  }
}

extern "C" void kernel_launch(void* const* d_in, const int* in_sizes, int n_in,
                              void* d_out, int out_size, void* d_ws, size_t ws_size,
                              hipStream_t stream) {
  const float* x0   = (const float*)d_in[0];
  const float* W1   = (const float*)d_in[1];
  const float* u1   = (const float*)d_in[2];
  const float* b1   = (const float*)d_in[3];
  const float* W2   = (const float*)d_in[4];
  const float* b2   = (const float*)d_in[5];
  const float* W3   = (const float*)d_in[6];
  const float* b3   = (const float*)d_in[7];
  const float* prec = (const float*)d_in[8];
  float* P = (float*)d_ws;   // H*H floats = 64 KB scratch

  vode_prep<<<(H * H) / 256, 256, 0, stream>>>(W1, W2, W3, P);
  vode_main<<<BATCH / MT, 256, 0, stream>>>(x0, W1, u1, b1, W2, b2, W3, b3,
                                            prec, P, (float*)d_out);
}